// AlpacaMH_15109694948018
// MI455X (gfx1250) — compile-verified
//
#include <hip/hip_runtime.h>
#include <cmath>

typedef float v2f __attribute__((ext_vector_type(2)));
typedef float v4f __attribute__((ext_vector_type(4)));
typedef float v8f __attribute__((ext_vector_type(8)));

// ---------------------------------------------------------------------------
// WMMA fp32 GEMM + bias (+ optional ELU) for the MLP stack.
// One wave (32 threads) computes a 16x16 tile of C = A(MxK) @ W(KxN) + b.
// V_WMMA_F32_16X16X4_F32 fragment layouts (ISA 7.12.2):
//   A 16x4 : lane<16 -> row M=lane,   {V0,V1}={K,K+1}; lane>=16 -> {K+2,K+3}
//   B 4x16 : lane<16 -> col N=lane,   {V0,V1}=rows {K,K+1}; lane>=16 -> {K+2,K+3}
//   C/D    : VGPR r -> row M = r + 8*(lane>>4), col N = lane&15
// ---------------------------------------------------------------------------
template <int K, bool DO_ELU>
__global__ __launch_bounds__(32)
void gemm_wmma_elu(const float* __restrict__ A, const float* __restrict__ W,
                   const float* __restrict__ bias, float* __restrict__ C, int N) {
    const int m0   = blockIdx.x << 4;
    const int n0   = blockIdx.y << 4;
    const int lane = threadIdx.x & 31;
    const int half = lane >> 4;
    const int l15  = lane & 15;
    const int n    = n0 + l15;

    const float* Arow = A + (m0 + l15) * K;

    v8f acc = {};
#pragma unroll 4
    for (int k = 0; k < K; k += 4) {
        // contiguous pair of A elements for this lane
        v2f a = *(const v2f*)(Arow + k + 2 * half);
        // two strided rows of W for this lane's column
        v2f b;
        b.x = W[(k + 2 * half + 0) * N + n];
        b.y = W[(k + 2 * half + 1) * N + n];
        acc = __builtin_amdgcn_wmma_f32_16x16x4_f32(
            /*neg_a=*/false, a, /*neg_b=*/false, b,
            /*c_mod=*/(short)0, acc, /*reuse_a=*/false, /*reuse_b=*/false);
    }

    const float bn = bias[n];
#pragma unroll
    for (int r = 0; r < 8; ++r) {
        float v = acc[r] + bn;
        if (DO_ELU) v = (v > 0.f) ? v : expm1f(v);   // jax.nn.elu, alpha=1
        C[(m0 + r + 8 * half) * N + n] = v;
    }
}

// ---------------------------------------------------------------------------
// Memory-bound fused bilinear pass: for each (b,z,u) tile (4096 total),
//   w_j   = sum_i phi_b[i] * Linv[i][j]        (1 FMA / 4 bytes streamed)
//   mu    = sum_j w_j * Qv[j]
//   sigma = 1 + sum_j w_j * phi_b[j]
//   out_cov = exp(logSigEps[u]) * sigma
// Block = 128 threads = 4 waves; wave g owns rows [32g, 32g+32); lane l owns
// columns 4l..4l+3 -> each row is one fully-coalesced 512B block load.
// Non-temporal loads: Linv (256MB) is read once; keep it out of L2.
// ---------------------------------------------------------------------------
__global__ __launch_bounds__(128)
void bilinear_stream(const float* __restrict__ Linv, const float* __restrict__ Q,
                     const float* __restrict__ phi,
                     const float* __restrict__ logSigEps,
                     float* __restrict__ out) {
    __shared__ float sPhi[128];
    __shared__ v4f   sW4[4][32];
    __shared__ float rA[128];
    __shared__ float rB[128];

    const int tid = threadIdx.x;
    const int g   = tid >> 5;   // wave id = row group
    const int l   = tid & 31;   // lane    = column group
    const int bid = blockIdx.x; // (b*Z + z)*U + u
    const int b   = bid >> 6;
    const int u   = bid & 7;

    sPhi[tid] = phi[(b << 7) + tid];
    __syncthreads();

    const float* T = Linv + ((size_t)bid << 14) + (size_t)(g << 5) * 128 + (l << 2);

    v4f w4 = {};
#pragma unroll 8
    for (int i = 0; i < 32; ++i) {
        const float p = sPhi[(g << 5) + i];              // LDS broadcast per wave
        v4f t = __builtin_nontemporal_load((const v4f*)(T + i * 128));
        w4.x = fmaf(p, t.x, w4.x);
        w4.y = fmaf(p, t.y, w4.y);
        w4.z = fmaf(p, t.z, w4.z);
        w4.w = fmaf(p, t.w, w4.w);
    }
    sW4[g][l] = w4;
    __syncthreads();

    // combine the 4 wave-partials for column c = tid, then the two dots
    const float* sWf = (const float*)sW4;
    const int   c  = tid;
    const float w  = sWf[c] + sWf[128 + c] + sWf[256 + c] + sWf[384 + c];
    const float qv = Q[((size_t)bid << 7) + c];

    rA[tid] = w * qv;       // -> mu
    rB[tid] = w * sPhi[c];  // -> sigma - 1
    __syncthreads();
#pragma unroll
    for (int s = 64; s > 0; s >>= 1) {
        if (tid < s) { rA[tid] += rA[tid + s]; rB[tid] += rB[tid + s]; }
        __syncthreads();
    }

    if (tid == 0) {
        out[bid]        = rA[0];                                  // mu (B,Z,U,1)
        out[4096 + bid] = expf(logSigEps[u]) * (1.f + rB[0]);     // pred_cov_diag
    }
}

// ---------------------------------------------------------------------------
// Harness entry point.
// Inputs (setup_inputs order):
//  0:x(64,64) 1:Linv(64,8,8,128,128) 2:Q(64,8,8,1,128)
//  3:W1(64,512) 4:b1(512) 5:W2(512,512) 6:b2(512)
//  7:W3(512,512) 8:b3(512) 9:W4(512,128) 10:b4(128) 11:logSigEps(8)
// Output: mu (4096 floats) ++ pred_cov_diag (4096 floats)
// ---------------------------------------------------------------------------
extern "C" void kernel_launch(void* const* d_in, const int* in_sizes, int n_in,
                              void* d_out, int out_size, void* d_ws, size_t ws_size,
                              hipStream_t stream) {
    const float* x         = (const float*)d_in[0];
    const float* Linv      = (const float*)d_in[1];
    const float* Q         = (const float*)d_in[2];
    const float* W1        = (const float*)d_in[3];
    const float* b1        = (const float*)d_in[4];
    const float* W2        = (const float*)d_in[5];
    const float* b2        = (const float*)d_in[6];
    const float* W3        = (const float*)d_in[7];
    const float* b3        = (const float*)d_in[8];
    const float* W4        = (const float*)d_in[9];
    const float* b4        = (const float*)d_in[10];
    const float* logSigEps = (const float*)d_in[11];
    float*       out       = (float*)d_out;

    // workspace layout (fp32): h1/h2/h3 (64x512 each), phi (64x128) = ~416 KB
    float* h1  = (float*)d_ws;
    float* h2  = h1 + 64 * 512;
    float* h3  = h2 + 64 * 512;
    float* phi = h3 + 64 * 512;

    const dim3 wave(32);
    gemm_wmma_elu<64,  true ><<<dim3(4, 32), wave, 0, stream>>>(x,  W1, b1, h1, 512);
    gemm_wmma_elu<512, true ><<<dim3(4, 32), wave, 0, stream>>>(h1, W2, b2, h2, 512);
    gemm_wmma_elu<512, true ><<<dim3(4, 32), wave, 0, stream>>>(h2, W3, b3, h3, 512);
    gemm_wmma_elu<512, false><<<dim3(4, 8),  wave, 0, stream>>>(h3, W4, b4, phi, 128);

    bilinear_stream<<<4096, 128, 0, stream>>>(Linv, Q, phi, logSigEps, out);
}